// RoiPoolingConv_41051297415443
// MI455X (gfx1250) — compile-verified
//
#include <hip/hip_runtime.h>

typedef __attribute__((ext_vector_type(2))) float v2f;
typedef __attribute__((ext_vector_type(8))) float v8f;

#define POOL 7
#define NROI 64
#define DIM  128
#define CH   32

// Per-axis sampling coords, identical math to the reference:
// c = start + (slot+0.5)*size/POOL - 0.5, clipped to [0, DIM-1]
__device__ __forceinline__ void axis_coords(float start, float size, int slot,
                                            int& i0, int& i1, float& w) {
  float c = start + ((float)slot + 0.5f) * (size * (1.0f / POOL)) - 0.5f;
  c = fminf(fmaxf(c, 0.0f), (float)(DIM - 1));
  float f = floorf(c);
  i0 = (int)f;
  i1 = min(i0 + 1, DIM - 1);
  w = c - f;
}

// One wave per output column (roi, ix, iy): computes out[roi, ix, iy, 0..6, 0..31]
// as a 16x56 (A=weights) x 56x32 (B=gathered corners) GEMM via
// V_WMMA_F32_16X16X4_F32, accumulated over 14 K-tiles x 2 channel halves.
__global__ void __launch_bounds__(256)
roi_align3d_wmma(const float* __restrict__ vol, const int* __restrict__ rois,
                 float* __restrict__ out) {
  const int lane = threadIdx.x & 31;
  const int wave = blockIdx.x * (blockDim.x >> 5) + (threadIdx.x >> 5);

  const int roi = wave / 49;
  const int col = wave - roi * 49;
  const int ix  = col / 7;
  const int iy  = col - ix * 7;

  const int* r = rois + roi * 6;
  const float sx = (float)r[0], sy = (float)r[1], sz = (float)r[2];
  const float ex = (float)r[3], ey = (float)r[4], ez = (float)r[5];

  int x0, x1, y0, y1;
  float wx, wy;
  axis_coords(sx, ex, ix, x0, x1, wx);
  axis_coords(sy, ey, iy, y0, y1, wy);

  const int m = lane & 15;   // A-row (M) / B-column (N) owned by this lane
  const int h = lane >> 4;   // K-half within the 4-wide K tile

  const float zstep = ez * (1.0f / POOL);
  const float zoff  = sz - 0.5f;

  v8f accLo = {};  // channels 0..15
  v8f accHi = {};  // channels 16..31

#pragma unroll
  for (int t = 0; t < 14; ++t) {
    // This lane supplies K = kA and K = kA+1 of tile t, kA = 4t + 2h.
    const int q  = 2 * t + h;       // 0..27
    const int ab = q & 3;           // xy corner: bit0 -> x1, bit1 -> y1
    const int iz = q >> 2;          // z output slot this K-pair belongs to

    // z coords for this K-pair (kA -> z0, kA+1 -> z1)
    float cz = fmaf((float)iz + 0.5f, zstep, zoff);
    cz = fminf(fmaxf(cz, 0.0f), (float)(DIM - 1));
    float fz = floorf(cz);
    int z0 = (int)fz;
    int z1 = min(z0 + 1, DIM - 1);
    float wz = cz - fz;

    const int   xs  = (ab & 1) ? x1 : x0;
    const int   ys  = (ab & 2) ? y1 : y0;
    const float wxa = (ab & 1) ? wx : 1.0f - wx;
    const float wyb = (ab & 2) ? wy : 1.0f - wy;
    const float wxy = wxa * wyb;

    // Gather B[kA, m(+16)] and B[kA+1, m(+16)]:
    // lanes 0..15 cover one 64B segment, lanes 16..31 another -> coalesced.
    const float* p = vol + (size_t)((xs * DIM + ys) * DIM) * CH + m;
    float b00 = p[(size_t)z0 * CH];        // K=kA,   N=m   (ch m)
    float b10 = p[(size_t)z1 * CH];        // K=kA+1, N=m
    float b01 = p[(size_t)z0 * CH + 16];   // K=kA,   ch m+16
    float b11 = p[(size_t)z1 * CH + 16];   // K=kA+1, ch m+16

    // A[m, kA] / A[m, kA+1]: nonzero only when this K-pair's iz == m.
    const float asel = (iz == m) ? wxy : 0.0f;
    v2f A   = { asel * (1.0f - wz), asel * wz };
    v2f BLo = { b00, b10 };
    v2f BHi = { b01, b11 };

    // 8 args: (neg_a, A, neg_b, B, c_mod, C, reuse_a, reuse_b)
    accLo = __builtin_amdgcn_wmma_f32_16x16x4_f32(
        false, A, false, BLo, (short)0, accLo, false, false);
    accHi = __builtin_amdgcn_wmma_f32_16x16x4_f32(
        false, A, false, BHi, (short)0, accHi, false, false);
  }

  // D layout: VGPR j, lanes 0..15 hold row M=j (our iz=j), N=lane.
  // Fold channels 16..31 (accHi, lanes 0..15) into lanes 16..31 for
  // full-wave 128B stores.
  float* o = out + (size_t)(((roi * 7 + ix) * 7 + iy) * 7) * CH + lane;
  const int src = (lane >= 16) ? (lane - 16) : lane;
#pragma unroll
  for (int j = 0; j < 7; ++j) {
    float hiv = __shfl(accHi[j], src, 32);
    float v   = (lane < 16) ? accLo[j] : hiv;
    o[(size_t)j * CH] = v;
  }
}

extern "C" void kernel_launch(void* const* d_in, const int* in_sizes, int n_in,
                              void* d_out, int out_size, void* d_ws, size_t ws_size,
                              hipStream_t stream) {
  (void)in_sizes; (void)n_in; (void)out_size; (void)d_ws; (void)ws_size;
  const float* vol  = (const float*)d_in[0];
  const int*   rois = (const int*)d_in[1];
  float*       out  = (float*)d_out;

  // 64 ROIs * 49 columns = 3136 waves = 392 blocks of 256 threads (no tail,
  // EXEC stays all-ones through the WMMAs).
  dim3 block(256);
  dim3 grid((NROI * 49 * 32) / 256);
  roi_align3d_wmma<<<grid, block, 0, stream>>>(vol, rois, out);
}